// RNNModel_74139725463871
// MI455X (gfx1250) — compile-verified
//
#include <hip/hip_runtime.h>
#include <hip/hip_bf16.h>

typedef __attribute__((ext_vector_type(2))) float v2f;
typedef __attribute__((ext_vector_type(8))) float v8f;

#define NHID 512
#define LDW  1024
#define TBE  (128 * 512)   // elements per (B, NHID) slab
#define KB   128           // K rows staged per LDS stage
#define STG_ELEMS (2 * KB * 16)   // 4096 floats per stage (c-tile + h-tile)

// ---------------------------------------------------------------------------
// One K=512 segment of the two 16x16 f32 WMMA chains (c-half / h-half), with
// B broadcast through LDS (all 8 waves of the block share the same n-tile).
//
// A fragment (16x4 f32): lanes 0-15 & 16-31 hold rows M=0..15, K pairs
//   klo,klo+1 (klo = (lane>=16)?2:0) -> one aligned float2 global load.
// B fragment: LDS laid out so lane reads rows klo,klo+1 at its column as one
//   aligned ds_load_b64:  idx(kk,c) = ((g*2+h2)*16 + c)*2 + r,
//   where g=kk>>2, h2=(kk>>1)&1, r=kk&1.
// ---------------------------------------------------------------------------
__device__ __forceinline__ void gemm512_lds(const float* __restrict__ A,
                                            const float* __restrict__ W,  // + n0
                                            float* __restrict__ lds,      // [2][4096]
                                            v8f& accC, v8f& accH,
                                            int tid, int l16, int half, int klo,
                                            int m0) {
    const int lc = tid & 15;        // loader column within tile
    const int lk = tid >> 4;        // loader base k-row (0..15)
    const int glk = lk >> 2, h2 = (lk >> 1) & 1, rr = lk & 1;
    const int i0 = ((glk * 2 + h2) * 16 + lc) * 2 + rr;   // +256 per 16 k-rows
    const float* arow = A + (size_t)(m0 + l16) * NHID + klo;

    // prolog: stage 0 -> buffer 0
    {
#pragma unroll
        for (int j = 0; j < 16; ++j) {
            const int tile = j >> 3, jj = j & 7;
            lds[tile * (KB * 16) + i0 + 256 * jj] =
                W[(size_t)(lk + 16 * jj) * LDW + tile * NHID + lc];
        }
    }
    __syncthreads();

#pragma unroll
    for (int s = 0; s < 4; ++s) {
        float pre[16];
        if (s < 3) {                 // prefetch next stage into registers
            const float* Wp = W + (size_t)(s + 1) * KB * LDW;
#pragma unroll
            for (int j = 0; j < 16; ++j) {
                const int tile = j >> 3, jj = j & 7;
                pre[j] = Wp[(size_t)(lk + 16 * jj) * LDW + tile * NHID + lc];
            }
        }

        const float* sb = lds + (s & 1) * STG_ELEMS;
        const float* ap = arow + s * KB;
#pragma unroll 8
        for (int g = 0; g < KB / 4; ++g) {
            v2f a = *(const v2f*)(ap + 4 * g);
            const float* bb = sb + ((g * 2 + half) * 16 + l16) * 2;
            v2f bc = *(const v2f*)bb;
            v2f bh = *(const v2f*)(bb + KB * 16);
            accC = __builtin_amdgcn_wmma_f32_16x16x4_f32(false, a, false, bc,
                                                         (short)0, accC, false, false);
            accH = __builtin_amdgcn_wmma_f32_16x16x4_f32(false, a, false, bh,
                                                         (short)0, accH, false, false);
        }

        if (s < 3) {
            __syncthreads();         // everyone done reading the other buffer
            float* db = lds + ((s + 1) & 1) * STG_ELEMS;
#pragma unroll
            for (int j = 0; j < 16; ++j) {
                const int tile = j >> 3, jj = j & 7;
                db[tile * (KB * 16) + i0 + 256 * jj] = pre[j];
            }
            __syncthreads();         // stores visible before next consume
        }
    }
}

// s = sp + sigmoid(c) * (act(h) - sp), using the 16x16 f32 C/D layout:
// VGPR r: lanes 0-15 -> M=m0+r, lanes 16-31 -> M=m0+8+r; N = n0 + (lane&15).
__device__ __forceinline__ void epilogue(const v8f& accC, const v8f& accH,
                                         const float* __restrict__ sp,
                                         float* __restrict__ sout, int act,
                                         int m0, int n0, int l16, int half) {
    const int col = n0 + l16;
#pragma unroll
    for (int r = 0; r < 8; ++r) {
        const int mrow = m0 + r + half * 8;
        const float c  = accC[r];
        const float hh = accH[r];
        const float spv = sp[mrow * NHID + col];
        const float g = 1.0f / (1.0f + expf(-c));
        float av;
        if (act == 0)      av = tanhf(hh);
        else if (act == 1) av = fmaxf(hh, 0.0f);
        else if (act == 2) av = 1.0f / (1.0f + expf(-hh));
        else               av = hh;
        sout[mrow * NHID + col] = spv + g * (av - spv);
    }
}

// ---------------------------------------------------------------------------
// Timestep entry cell: ch = concat(x, h) @ W0 (K=1024), s0 = h + sig(c)*(tanh(h0)-h)
// ---------------------------------------------------------------------------
__global__ __launch_bounds__(256) void darts_cell0(const float* __restrict__ x,
                                                   const float* __restrict__ h,
                                                   const float* __restrict__ W0,
                                                   float* __restrict__ sout) {
    __shared__ float ldsB[2 * STG_ELEMS];
    const int tid = threadIdx.x;
    const int wave = tid >> 5, lane = tid & 31;
    const int half = lane >> 4, l16 = lane & 15, klo = half * 2;
    const int m0 = wave * 16;            // 8 waves = 8 row tiles
    const int n0 = blockIdx.x * 16;      // 32 blocks = 32 state-column tiles

    v8f accC = {};
    v8f accH = {};
    // W0 rows 0..511 multiply x
    gemm512_lds(x, W0 + n0, ldsB, accC, accH, tid, l16, half, klo, m0);
    __syncthreads();
    // W0 rows 512..1023 multiply h_prev
    gemm512_lds(h, W0 + (size_t)NHID * LDW + n0, ldsB, accC, accH,
                tid, l16, half, klo, m0);

    epilogue(accC, accH, h, sout, /*tanh*/ 0, m0, n0, l16, half);
}

// ---------------------------------------------------------------------------
// Batched DAG step: up to 3 independent GEMM+gate updates, one per blockIdx.y
// ---------------------------------------------------------------------------
struct Sub {
    const float* sp;
    const float* W;
    float* sout;
    int act;
};
struct Subs {
    Sub s[3];
};

__global__ __launch_bounds__(256) void darts_cell_step(Subs subs) {
    __shared__ float ldsB[2 * STG_ELEMS];
    const Sub sb = subs.s[blockIdx.y];   // uniform per block
    const int tid = threadIdx.x;
    const int wave = tid >> 5, lane = tid & 31;
    const int half = lane >> 4, l16 = lane & 15, klo = half * 2;
    const int m0 = wave * 16;
    const int n0 = blockIdx.x * 16;

    v8f accC = {};
    v8f accH = {};
    gemm512_lds(sb.sp, sb.W + n0, ldsB, accC, accH, tid, l16, half, klo, m0);

    epilogue(accC, accH, sb.sp, sb.sout, sb.act, m0, n0, l16, half);
}

// ---------------------------------------------------------------------------
// h_new = mean(states[1..8]); optionally duplicate to the trailing output slab
// ---------------------------------------------------------------------------
__global__ __launch_bounds__(256) void darts_mean(const float* __restrict__ S,
                                                  float* __restrict__ out,
                                                  float* __restrict__ out2) {
    const int i = blockIdx.x * blockDim.x + threadIdx.x;   // 65536 elements
    float acc = 0.0f;
#pragma unroll
    for (int j = 1; j <= 8; ++j) acc += S[(size_t)j * TBE + i];
    const float m = acc * 0.125f;
    out[i] = m;
    if (out2) out2[i] = m;
}

// ---------------------------------------------------------------------------

extern "C" void kernel_launch(void* const* d_in, const int* in_sizes, int n_in,
                              void* d_out, int out_size, void* d_ws, size_t ws_size,
                              hipStream_t stream) {
    const float* inputs = (const float*)d_in[0];   // (256,128,512)
    const float* hidden = (const float*)d_in[1];   // (1,128,512) zeros
    const float* W0     = (const float*)d_in[2];   // (1024,1024)
    const float* WsB    = (const float*)d_in[3];   // (8,512,1024)
    float* out = (float*)d_out;
    float* S   = (float*)d_ws;                     // 9 state slabs: 2.36 MB

    static const int PRED[8] = {0, 1, 1, 1, 2, 5, 3, 5};
    // act codes: 0=tanh 1=relu 2=sigmoid 3=identity
    static const int ACTC[8] = {2, 1, 1, 3, 0, 2, 0, 1};

    const dim3 blk(256);
    auto Sb = [&](int i) { return S + (size_t)i * TBE; };
    auto Wi = [&](int i) { return WsB + (size_t)i * (size_t)NHID * LDW; };
    auto mk = [&](int j) {
        Sub s;
        s.sp = Sb(PRED[j]);
        s.W = Wi(j);
        s.sout = Sb(j + 1);
        s.act = ACTC[j];
        return s;
    };

    for (int t = 0; t < 256; ++t) {
        const float* x = inputs + (size_t)t * TBE;
        const float* h = (t == 0) ? hidden : out + (size_t)(t - 1) * TBE;

        // states[0]
        darts_cell0<<<dim3(32), blk, 0, stream>>>(x, h, W0, Sb(0));

        // step 0 -> states[1]
        Subs g1; g1.s[0] = mk(0); g1.s[1] = g1.s[0]; g1.s[2] = g1.s[0];
        darts_cell_step<<<dim3(32, 1), blk, 0, stream>>>(g1);

        // steps 1,2,3 (all pred states[1]) -> states[2..4]
        Subs g2; g2.s[0] = mk(1); g2.s[1] = mk(2); g2.s[2] = mk(3);
        darts_cell_step<<<dim3(32, 3), blk, 0, stream>>>(g2);

        // steps 4 (pred states[2]) and 6 (pred states[3]) -> states[5], states[7]
        Subs g3; g3.s[0] = mk(4); g3.s[1] = mk(6); g3.s[2] = g3.s[0];
        darts_cell_step<<<dim3(32, 2), blk, 0, stream>>>(g3);

        // steps 5,7 (pred states[5]) -> states[6], states[8]
        Subs g4; g4.s[0] = mk(5); g4.s[1] = mk(7); g4.s[2] = g4.s[0];
        darts_cell_step<<<dim3(32, 2), blk, 0, stream>>>(g4);

        // hiddens[t] = mean(states[1..8]); t==255 also fills the trailing slab
        float* o2 = (t == 255) ? (out + (size_t)256 * TBE) : (float*)nullptr;
        darts_mean<<<dim3(256), blk, 0, stream>>>(S, out + (size_t)t * TBE, o2);
    }
    (void)in_sizes; (void)n_in; (void)out_size; (void)ws_size;
}